// LocalFeatureAggregation_18631568130515
// MI455X (gfx1250) — compile-verified
//
#include <hip/hip_runtime.h>

// LocalFeatureAggregation (RandLA-Net style) for MI455X / gfx1250.
// One wave32 per point; per-point 16xK GEMMs on V_WMMA_F32_16X16X4_F32.
// Attention tile for broadcast columns is folded out analytically
// (softmax sums to 1 -> pooled[c>=16] == x[c]), so only 11 WMMAs per pool.
#define BB   2
#define NN   65536
#define KK   16
#define DIN  8
#define DOUT 32
#define EPSV 1e-6f
#define XK_STRIDE 34   // 16x32 tile padded: 34*4B, 17 coprime 32 -> conflict-free rows

typedef float v2f __attribute__((ext_vector_type(2)));
typedef float v8f __attribute__((ext_vector_type(8)));

static __device__ __forceinline__ v8f wmma4(v2f a, v2f b, v8f c) {
  // D = A(16x4,f32) * B(4x16,f32) + C(16x16,f32)
  return __builtin_amdgcn_wmma_f32_16x16x4_f32(false, a, false, b, (short)0, c, false, false);
}

static __device__ __forceinline__ void lds_fence() {
  // same-wave LDS ops are in-order; this guarantees cross-lane visibility
  asm volatile("s_wait_dscnt 0" ::: "memory");
}

struct Params {
  const float *W1, *b1;
  const float *Wl1, *bl1, *gl1, *bel1, *Ws1, *Wm1, *bm1, *gm1, *bem1;
  const float *Wl2, *bl2, *gl2, *bel2, *Ws2, *Wm2, *bm2, *gm2, *bem2;
  const float *W2, *b2, *Wsc, *bsc, *gsc, *besc;
};

// One LSE + attentive-pool stage. Returns the per-lane output channel value:
// channel = row (outc==16, duplicated over lane halves) or = lane (outc==32).
static __device__ __forceinline__ float lse_pool(
    const float g[12], float xv, int outc,
    const float* __restrict__ Wl, const float* __restrict__ bl,
    const float* __restrict__ gl, const float* __restrict__ bel,
    const float* __restrict__ Ws, const float* __restrict__ Wm,
    const float* __restrict__ bm, const float* __restrict__ gm,
    const float* __restrict__ bem,
    float* XKw, float* VECw, int lane, int row, int hi)
{
  const float inv = rsqrtf(1.0f + EPSV);
  const bool  hb  = (hi != 0);

  // ---- sp = relu(bn(geo @ Wl^T + bl)) : 16x10 @ 10x16, K padded 10->12 -> 3 WMMAs.
  // A/B frag layout (f32 K=4): VGPR v, lane-half hi -> K = 4s + 2*hi + v.
  // All selects use compile-time indices -> single v_cndmask each, no exec branches.
  v8f acc = {};
#pragma unroll
  for (int s = 0; s < 3; ++s) {
    v2f a, b;
    a[0] = hb ? g[4 * s + 2] : g[4 * s + 0];
    a[1] = hb ? g[4 * s + 3] : g[4 * s + 1];
    if (s < 2) {
      b[0] = Wl[row * 10 + 4 * s + 2 * hi];
      b[1] = Wl[row * 10 + 4 * s + 2 * hi + 1];
    } else {
      // k0 = 8 + 2*hi: hi half hits padded K (10,11) -> zero; load in-bounds, select.
      const float w8 = Wl[row * 10 + 8];
      const float w9 = Wl[row * 10 + 9];
      b[0] = hb ? 0.0f : w8;
      b[1] = hb ? 0.0f : w9;
    }
    acc = wmma4(a, b, acc);
  }
  const float blv = bl[row], sclv = gl[row] * inv, bev = bel[row];
  float sp[8];
#pragma unroll
  for (int j = 0; j < 8; ++j)
    sp[j] = fmaxf((acc[j] + blv) * sclv + bev, 0.0f);

  // ---- stage xk = [sp | broadcast(x)] (16x32) row-major into LDS (C-layout -> rows)
#pragma unroll
  for (int j = 0; j < 8; ++j) {
    const int m = j + 8 * hi;
    XKw[m * XK_STRIDE + row]      = sp[j];   // columns 0..15
    XKw[m * XK_STRIDE + 16 + row] = xv;      // columns 16..31: x broadcast over rows
  }
  lds_fence();

  // ---- scores tile 0 only: columns >=16 of xk are constant over K, so their
  // pooled value is exactly x (softmax weights sum to 1) -- no WMMA needed.
  v8f sc = {};
#pragma unroll
  for (int s = 0; s < 8; ++s) {
    const int k0 = 4 * s + 2 * hi;
    const v2f a = *(const v2f*)(XKw + row * XK_STRIDE + k0);  // 8B-aligned ds_load_b64
    v2f b;
    b[0] = Ws[row * 32 + k0];        // Ws^T[k][row], output channels 0..15
    b[1] = Ws[row * 32 + k0 + 1];
    sc = wmma4(a, b, sc);
  }

  // column softmax over the 16 neighbor rows: 8 regs here + 8 in lane^16
  float mx = sc[0];
#pragma unroll
  for (int j = 1; j < 8; ++j) mx = fmaxf(mx, sc[j]);
  mx = fmaxf(mx, __shfl_xor(mx, 16, 32));
  float e[8], sum = 0.0f;
#pragma unroll
  for (int j = 0; j < 8; ++j) { e[j] = __expf(sc[j] - mx); sum += e[j]; }
  sum += __shfl_xor(sum, 16, 32);

  float p = 0.0f;
#pragma unroll
  for (int j = 0; j < 8; ++j) p += e[j] * sp[j];
  p += __shfl_xor(p, 16, 32);
  p *= (1.0f / sum);                 // pooled[row], exact weighted mean

  // ---- broadcast pooled (32 values: [attn-pooled sp | x]) and apply Wm head
  VECw[row]      = p;                // lane halves write identical values
  VECw[16 + row] = xv;               // pooled[c>=16] == x[c] analytically
  lds_fence();

  const int nc = (outc == 16) ? row : lane;
  float o = 0.0f;
#pragma unroll
  for (int k = 0; k < 32; ++k) o += VECw[k] * Wm[nc * 32 + k];
  o = fmaxf((o + bm[nc]) * (gm[nc] * inv) + bem[nc], 0.0f);
  lds_fence();
  return o;
}

__global__ void __launch_bounds__(256) lfa_kernel(
    const float* __restrict__ coords, const float* __restrict__ features,
    const int* __restrict__ idx, const float* __restrict__ dist,
    Params prm, float* __restrict__ out)
{
  __shared__ float XK[8][16 * XK_STRIDE];
  __shared__ float VEC[8][32];

  const int tid  = blockIdx.x * blockDim.x + threadIdx.x;
  const int wid  = tid >> 5;              // global wave id == point id
  const int lane = threadIdx.x & 31;
  const int wv   = threadIdx.x >> 5;
  const int row  = lane & 15;
  const int hi   = lane >> 4;

  const int b = wid / NN;
  const int n = wid - b * NN;

  // point features, channel-major: features[b][c][n]
  float feats[DIN];
#pragma unroll
  for (int c = 0; c < DIN; ++c)
    feats[c] = features[(b * DIN + c) * NN + n];

  // mlp1: x = leaky_relu(feats @ W1^T + b1, 0.2); lane holds channel `row`
  float xv = prm.b1[row];
#pragma unroll
  for (int c = 0; c < DIN; ++c) xv += feats[c] * prm.W1[row * DIN + c];
  xv = fmaxf(xv, 0.2f * xv);

  // geometric features for neighbor `row`: [center, nbr, center-nbr, dist], pad to 12
  float g[12];
  const int nb = idx[(b * NN + n) * KK + row];
#pragma unroll
  for (int c = 0; c < 3; ++c) {
    const float ce = coords[(b * NN + n)  * 3 + c];
    const float nv = coords[(b * NN + nb) * 3 + c];
    g[c] = ce; g[3 + c] = nv; g[6 + c] = ce - nv;
  }
  g[9]  = dist[(b * NN + n) * KK + row];
  g[10] = 0.0f; g[11] = 0.0f;

  float* XKw  = XK[wv];
  float* VECw = VEC[wv];

  xv = lse_pool(g, xv, 16, prm.Wl1, prm.bl1, prm.gl1, prm.bel1,
                prm.Ws1, prm.Wm1, prm.bm1, prm.gm1, prm.bem1,
                XKw, VECw, lane, row, hi);
  const float x2 = lse_pool(g, xv, 32, prm.Wl2, prm.bl2, prm.gl2, prm.bel2,
                prm.Ws2, prm.Wm2, prm.bm2, prm.gm2, prm.bem2,
                XKw, VECw, lane, row, hi);

  // broadcast x2 (32 channels), then mlp2 + BN shortcut + leaky_relu(0.01)
  VECw[lane] = x2;
  lds_fence();

  const float inv = rsqrtf(1.0f + EPSV);
#pragma unroll
  for (int h = 0; h < 2; ++h) {
    const int ch = lane + 32 * h;
    float m = prm.b2[ch];
#pragma unroll
    for (int k = 0; k < 32; ++k) m += VECw[k] * prm.W2[ch * 32 + k];
    float s = prm.bsc[ch];
#pragma unroll
    for (int c = 0; c < DIN; ++c) s += feats[c] * prm.Wsc[ch * DIN + c];
    s = s * (prm.gsc[ch] * inv) + prm.besc[ch];
    float r = m + s;
    r = fmaxf(r, 0.01f * r);
    out[(b * 64 + ch) * NN + n] = r;
  }
}

extern "C" void kernel_launch(void* const* d_in, const int* in_sizes, int n_in,
                              void* d_out, int out_size, void* d_ws, size_t ws_size,
                              hipStream_t stream) {
  (void)in_sizes; (void)n_in; (void)d_ws; (void)ws_size; (void)out_size;
  const float* coords   = (const float*)d_in[0];
  const float* features = (const float*)d_in[1];
  const int*   idx      = (const int*)  d_in[2];
  const float* dist     = (const float*)d_in[3];

  Params p;
  int i = 4;
  p.W1  = (const float*)d_in[i++]; p.b1   = (const float*)d_in[i++];
  p.Wl1 = (const float*)d_in[i++]; p.bl1  = (const float*)d_in[i++];
  p.gl1 = (const float*)d_in[i++]; p.bel1 = (const float*)d_in[i++];
  p.Ws1 = (const float*)d_in[i++];
  p.Wm1 = (const float*)d_in[i++]; p.bm1  = (const float*)d_in[i++];
  p.gm1 = (const float*)d_in[i++]; p.bem1 = (const float*)d_in[i++];
  p.Wl2 = (const float*)d_in[i++]; p.bl2  = (const float*)d_in[i++];
  p.gl2 = (const float*)d_in[i++]; p.bel2 = (const float*)d_in[i++];
  p.Ws2 = (const float*)d_in[i++];
  p.Wm2 = (const float*)d_in[i++]; p.bm2  = (const float*)d_in[i++];
  p.gm2 = (const float*)d_in[i++]; p.bem2 = (const float*)d_in[i++];
  p.W2  = (const float*)d_in[i++]; p.b2   = (const float*)d_in[i++];
  p.Wsc = (const float*)d_in[i++]; p.bsc  = (const float*)d_in[i++];
  p.gsc = (const float*)d_in[i++]; p.besc = (const float*)d_in[i++];

  // one wave32 per point, exact grid so EXEC is all-1s everywhere (WMMA requirement)
  const long long total_threads = (long long)BB * NN * 32;  // 4,194,304
  dim3 block(256);
  dim3 grid((unsigned)(total_threads / 256));               // 16384 blocks
  lfa_kernel<<<grid, block, 0, stream>>>(coords, features, idx, dist, p, (float*)d_out);
}